// LocalModule_87909390614690
// MI455X (gfx1250) — compile-verified
//
#include <hip/hip_runtime.h>
#include <hip/hip_bf16.h>

typedef float v2f __attribute__((ext_vector_type(2)));
typedef float v8f __attribute__((ext_vector_type(8)));

__device__ __forceinline__ v8f wmma4(v2f a, v2f b, v8f c) {
    // D = A(16x4 f32) * B(4x16 f32) + C(16x16 f32)
    return __builtin_amdgcn_wmma_f32_16x16x4_f32(false, a, false, b, (short)0, c, false, false);
}

__device__ __forceinline__ v2f ld2(const float* p) {  // 8B-aligned pair load
    return *(const v2f*)p;
}

__device__ __forceinline__ v8f zero8() {
    v8f z;
#pragma unroll
    for (int i = 0; i < 8; ++i) z[i] = 0.f;
    return z;
}

__device__ __forceinline__ float sigf(float x) { return 1.f / (1.f + expf(-x)); }

// B staging layout helper: element (k, n) of a KxN B-matrix lives at
//   sB[((k>>1)*N + n)*2 + (k&1)]
// so the WMMA pair {B[k0][n], B[k0+1][n]} (k0 even) is one aligned b64.
#define BP(k, n, N) ((((k) >> 1) * (N) + (n)) * 2 + ((k) & 1))

// ---------------------------------------------------------------------------
// Stage 1: anchor grouping + top-8 nearest + feature gather
// grid = 512 frames, block = 128
// ---------------------------------------------------------------------------
__global__ void __launch_bounds__(128)
group_kernel(const float* __restrict__ x, float* __restrict__ gp)
{
    __shared__ float sXYZ[4096 * 3];
    __shared__ float sCd[128 * 8];
    __shared__ int   sCi[128 * 8];

    const int tid = threadIdx.x;
    const long frame = blockIdx.x;
    const float* xf = x + frame * 4096 * 28;

    for (int i = tid; i < 4096 * 3; i += 128) {
        int p = i / 3, c = i % 3;
        sXYZ[i] = xf[p * 28 + c];
    }
    __syncthreads();

    for (int a = 0; a < 27; ++a) {
        const float ax = 0.3f * (float)(a % 3);
        const float ay = 0.3f * (float)((a / 3) % 3) - 0.3f;
        const float az = 0.3f * (float)(a / 9) - 0.3f;
        const float an2 = ax * ax + ay * ay + az * az;

        float bd[8]; int bi[8];
#pragma unroll
        for (int s = 0; s < 8; ++s) { bd[s] = __builtin_inff(); bi[s] = 0x7fffffff; }

        for (int p = tid; p < 4096; p += 128) {
            float px = sXYZ[p * 3 + 0], py = sXYZ[p * 3 + 1], pz = sXYZ[p * 3 + 2];
            bool pad = (px == 0.f) && (py == 0.f) && (pz == 0.f);
            float d = pad ? __builtin_inff()
                          : (an2 + px * px + py * py + pz * pz
                             - 2.f * (ax * px + ay * py + az * pz));
            if (d < bd[7] || (d == bd[7] && p < bi[7])) {
                bd[7] = d; bi[7] = p;
#pragma unroll
                for (int s = 7; s > 0; --s) {
                    bool sw = (bd[s] < bd[s - 1]) || (bd[s] == bd[s - 1] && bi[s] < bi[s - 1]);
                    if (sw) {
                        float td = bd[s]; bd[s] = bd[s - 1]; bd[s - 1] = td;
                        int   ti = bi[s]; bi[s] = bi[s - 1]; bi[s - 1] = ti;
                    }
                }
            }
        }
#pragma unroll
        for (int s = 0; s < 8; ++s) { sCd[tid * 8 + s] = bd[s]; sCi[tid * 8 + s] = bi[s]; }
        __syncthreads();

        for (int stride = 64; stride >= 1; stride >>= 1) {
            if (tid < stride) {
                float md[8]; int mi[8];
                int ia = 0, ib = 0;
                float* Ad = &sCd[tid * 8];            int* Ai = &sCi[tid * 8];
                float* Bd = &sCd[(tid + stride) * 8]; int* Bi = &sCi[(tid + stride) * 8];
#pragma unroll
                for (int s = 0; s < 8; ++s) {
                    bool takeA = (Ad[ia] < Bd[ib]) || (Ad[ia] == Bd[ib] && Ai[ia] <= Bi[ib]);
                    if (takeA) { md[s] = Ad[ia]; mi[s] = Ai[ia]; ++ia; }
                    else       { md[s] = Bd[ib]; mi[s] = Bi[ib]; ++ib; }
                }
#pragma unroll
                for (int s = 0; s < 8; ++s) { Ad[s] = md[s]; Ai[s] = mi[s]; }
            }
            __syncthreads();
        }

        if (tid < 8) {
            int p = sCi[tid];
            if (p < 0 || p >= 4096) p = 0;
            float px = sXYZ[p * 3 + 0], py = sXYZ[p * 3 + 1], pz = sXYZ[p * 3 + 2];
            long row = ((frame * 27 + a) * 8 + tid) * 31;
            gp[row + 0] = ax; gp[row + 1] = ay; gp[row + 2] = az;
            gp[row + 3] = px - ax; gp[row + 4] = py - ay; gp[row + 5] = pz - az;
            for (int f = 0; f < 25; ++f) gp[row + 6 + f] = xf[p * 28 + 3 + f];
        }
        __syncthreads();
    }
}

// ---------------------------------------------------------------------------
// Stage 2: PointNet MLP (31->32->48->64) + attention pooling.  WMMA f32.
// grid = 13824/8 = 1728, block = 128 (4 waves, 2 anchors (16 rows) per wave)
// ---------------------------------------------------------------------------
__global__ void __launch_bounds__(128)
pointnet_kernel(const float* __restrict__ gp,
                const float* __restrict__ W1, const float* __restrict__ b1,
                const float* __restrict__ g1, const float* __restrict__ be1,
                const float* __restrict__ W2, const float* __restrict__ b2,
                const float* __restrict__ g2, const float* __restrict__ be2,
                const float* __restrict__ W3, const float* __restrict__ b3,
                const float* __restrict__ g3, const float* __restrict__ be3,
                const float* __restrict__ Wa, const float* __restrict__ ba,
                float* __restrict__ attn_out, float* __restrict__ vec_out)
{
    __shared__ float sW1[32 * 32];   // K-pair interleaved, K padded 31->32, N=32
    __shared__ float sW2[32 * 48];
    __shared__ float sW3[48 * 64];
    __shared__ float sWa[64];
    __shared__ float sBuf0[4][16 * 64];
    __shared__ float sBuf1[4][16 * 64];
    __shared__ float sLog[4][16];
    __shared__ float sAw[4][16];

    const int tid = threadIdx.x;
    const int wave = tid >> 5, lane = tid & 31;
    const int mrow = lane & 15;
    const int kb = (lane >> 4) << 1;      // 0 or 2
    const float bnr = rsqrtf(1.0f + 1e-5f);

    for (int i = tid; i < 32 * 32; i += 128) {
        int k = i / 32, n = i % 32;
        sW1[BP(k, n, 32)] = (k < 31) ? W1[n * 31 + k] : 0.f;
    }
    for (int i = tid; i < 32 * 48; i += 128) {
        int k = i / 48, n = i % 48;
        sW2[BP(k, n, 48)] = W2[n * 32 + k];
    }
    for (int i = tid; i < 48 * 64; i += 128) {
        int k = i / 64, n = i % 64;
        sW3[BP(k, n, 64)] = W3[n * 48 + k];
    }
    if (tid < 64) sWa[tid] = Wa[tid];
    __syncthreads();

    const long fa0 = (long)blockIdx.x * 8 + wave * 2;   // first frame-anchor of wave
    float* sA = sBuf0[wave];
    float* sB = sBuf1[wave];

    // load A rows: 16 rows x 31 feats, zero-padded to 32 (row stride 64)
    for (int i = lane; i < 16 * 64; i += 32) {
        int r = i >> 6, f = i & 63;
        float v = 0.f;
        if (f < 31) v = gp[((fa0 + (r >> 3)) * 8 + (r & 7)) * 31 + f];
        sA[i] = v;
    }

    // ---- layer1: K=32 (31 padded), N=32 ----
    for (int nt = 0; nt < 2; ++nt) {
        v8f c = zero8();
        for (int kk = 0; kk < 8; ++kk) {
            int k0 = kk * 4 + kb;
            v2f a = ld2(&sA[mrow * 64 + k0]);
            v2f b = ld2(&sW1[((k0 >> 1) * 32 + nt * 16 + mrow) * 2]);
            c = wmma4(a, b, c);
        }
        int n = nt * 16 + mrow;
        float sc = g1[n] * bnr, sh = be1[n], bs = b1[n];
#pragma unroll
        for (int i = 0; i < 8; ++i) {
            int m = i + ((lane >> 4) << 3);
            sB[m * 64 + n] = fmaxf((c[i] + bs) * sc + sh, 0.f);
        }
    }
    // ---- layer2: K=32, N=48 (in sB -> out sA) ----
    for (int nt = 0; nt < 3; ++nt) {
        v8f c = zero8();
        for (int kk = 0; kk < 8; ++kk) {
            int k0 = kk * 4 + kb;
            v2f a = ld2(&sB[mrow * 64 + k0]);
            v2f b = ld2(&sW2[((k0 >> 1) * 48 + nt * 16 + mrow) * 2]);
            c = wmma4(a, b, c);
        }
        int n = nt * 16 + mrow;
        float sc = g2[n] * bnr, sh = be2[n], bs = b2[n];
#pragma unroll
        for (int i = 0; i < 8; ++i) {
            int m = i + ((lane >> 4) << 3);
            sA[m * 64 + n] = fmaxf((c[i] + bs) * sc + sh, 0.f);
        }
    }
    // ---- layer3: K=48, N=64 (in sA -> out sB) ----
    for (int nt = 0; nt < 4; ++nt) {
        v8f c = zero8();
        for (int kk = 0; kk < 12; ++kk) {
            int k0 = kk * 4 + kb;
            v2f a = ld2(&sA[mrow * 64 + k0]);
            v2f b = ld2(&sW3[((k0 >> 1) * 64 + nt * 16 + mrow) * 2]);
            c = wmma4(a, b, c);
        }
        int n = nt * 16 + mrow;
        float sc = g3[n] * bnr, sh = be3[n], bs = b3[n];
#pragma unroll
        for (int i = 0; i < 8; ++i) {
            int m = i + ((lane >> 4) << 3);
            sB[m * 64 + n] = fmaxf((c[i] + bs) * sc + sh, 0.f);
        }
    }
    __syncthreads();

    // attention logits
    if (lane < 16) {
        float s = ba[0];
        for (int f = 0; f < 64; ++f) s += sB[lane * 64 + f] * sWa[f];
        sLog[wave][lane] = s;
    }
    __syncthreads();
    // softmax over 8 samples per anchor (2 anchors/wave)
    if (lane < 2) {
        int base = lane * 8;
        float mx = -3.4e38f;
        for (int s = 0; s < 8; ++s) mx = fmaxf(mx, sLog[wave][base + s]);
        float e[8], sum = 0.f;
        for (int s = 0; s < 8; ++s) { e[s] = expf(sLog[wave][base + s] - mx); sum += e[s]; }
        float inv = 1.f / sum;
        for (int s = 0; s < 8; ++s) {
            float w = e[s] * inv;
            sAw[wave][base + s] = w;
            attn_out[(fa0 + lane) * 8 + s] = w;
        }
    }
    __syncthreads();
    // weighted sum -> vec
    for (int i = lane; i < 128; i += 32) {
        int al = i >> 6, dd = i & 63;
        float acc = 0.f;
        for (int s = 0; s < 8; ++s) acc += sB[(al * 8 + s) * 64 + dd] * sAw[wave][al * 8 + s];
        vec_out[(fa0 + al) * 64 + dd] = acc;
    }
}

// ---------------------------------------------------------------------------
// Stage 3: voxel conv (GEMM 512x1728x96) + MLP 96->128->64.  WMMA f32.
// grid = 32 (16 frames each), block = 32 (one wave)
// ---------------------------------------------------------------------------
__global__ void __launch_bounds__(32)
voxel_kernel(const float* __restrict__ vec, const float* __restrict__ K1,
             const float* __restrict__ vb1, const float* __restrict__ vg1, const float* __restrict__ vbe1,
             const float* __restrict__ VW2, const float* __restrict__ vb2,
             const float* __restrict__ vg2, const float* __restrict__ vbe2,
             const float* __restrict__ VW3, const float* __restrict__ vb3,
             const float* __restrict__ vg3, const float* __restrict__ vbe3,
             float* __restrict__ seq)
{
    __shared__ float sB[6144];          // max chunk: 64x96 / 48x128 / 64x64 (pair layout)
    __shared__ float sIn[16 * 128];
    __shared__ float sOut[16 * 128];

    const int lane = threadIdx.x;
    const int mrow = lane & 15;
    const int kb = (lane >> 4) << 1;
    const int fr0 = blockIdx.x * 16;
    const float bnr = rsqrtf(1.0f + 1e-5f);

    v8f acc[8];
    // ---- voxel conv: K loop over 27 anchor chunks of 64 ----
    for (int t = 0; t < 6; ++t) acc[t] = zero8();
    for (int az = 0; az < 27; ++az) {
        for (int i = lane; i < 64 * 96; i += 32) {
            int k = i / 96, o = i % 96;
            sB[BP(k, o, 96)] = K1[(long)(o * 64 + k) * 27 + az];
        }
        for (int i = lane; i < 16 * 64; i += 32) {
            int r = i >> 6, c = i & 63;
            sIn[i] = vec[((long)(fr0 + r) * 27 + az) * 64 + c];
        }
        __syncthreads();
        for (int kk = 0; kk < 16; ++kk) {
            int k0 = kk * 4 + kb;
            v2f a = ld2(&sIn[mrow * 64 + k0]);
            for (int nt = 0; nt < 6; ++nt) {
                v2f b = ld2(&sB[((k0 >> 1) * 96 + nt * 16 + mrow) * 2]);
                acc[nt] = wmma4(a, b, acc[nt]);
            }
        }
        __syncthreads();
    }
    for (int nt = 0; nt < 6; ++nt) {
        int n = nt * 16 + mrow;
        float sc = vg1[n] * bnr, sh = vbe1[n], bs = vb1[n];
#pragma unroll
        for (int i = 0; i < 8; ++i) {
            int m = i + ((lane >> 4) << 3);
            sOut[m * 96 + n] = fmaxf((acc[nt][i] + bs) * sc + sh, 0.f);
        }
    }
    __syncthreads();
    // ---- layer2: K=96 (2 chunks of 48), N=128 ----
    for (int t = 0; t < 8; ++t) acc[t] = zero8();
    for (int kc = 0; kc < 2; ++kc) {
        for (int i = lane; i < 48 * 128; i += 32) {
            int k = i / 128, n = i % 128;
            sB[BP(k, n, 128)] = VW2[(long)n * 96 + kc * 48 + k];
        }
        __syncthreads();
        for (int kk = 0; kk < 12; ++kk) {
            int k0b = kk * 4 + kb;
            int k0a = kc * 48 + k0b;
            v2f a = ld2(&sOut[mrow * 96 + k0a]);
            for (int nt = 0; nt < 8; ++nt) {
                v2f b = ld2(&sB[((k0b >> 1) * 128 + nt * 16 + mrow) * 2]);
                acc[nt] = wmma4(a, b, acc[nt]);
            }
        }
        __syncthreads();
    }
    for (int nt = 0; nt < 8; ++nt) {
        int n = nt * 16 + mrow;
        float sc = vg2[n] * bnr, sh = vbe2[n], bs = vb2[n];
#pragma unroll
        for (int i = 0; i < 8; ++i) {
            int m = i + ((lane >> 4) << 3);
            sIn[m * 128 + n] = fmaxf((acc[nt][i] + bs) * sc + sh, 0.f);
        }
    }
    __syncthreads();
    // ---- layer3: K=128 (2 chunks of 64), N=64 ----
    for (int t = 0; t < 4; ++t) acc[t] = zero8();
    for (int kc = 0; kc < 2; ++kc) {
        for (int i = lane; i < 64 * 64; i += 32) {
            int k = i / 64, n = i % 64;
            sB[BP(k, n, 64)] = VW3[(long)n * 128 + kc * 64 + k];
        }
        __syncthreads();
        for (int kk = 0; kk < 16; ++kk) {
            int k0b = kk * 4 + kb;
            int k0a = kc * 64 + k0b;
            v2f a = ld2(&sIn[mrow * 128 + k0a]);
            for (int nt = 0; nt < 4; ++nt) {
                v2f b = ld2(&sB[((k0b >> 1) * 64 + nt * 16 + mrow) * 2]);
                acc[nt] = wmma4(a, b, acc[nt]);
            }
        }
        __syncthreads();
    }
    for (int nt = 0; nt < 4; ++nt) {
        int n = nt * 16 + mrow;
        float sc = vg3[n] * bnr, sh = vbe3[n], bs = vb3[n];
#pragma unroll
        for (int i = 0; i < 8; ++i) {
            int m = i + ((lane >> 4) << 3);
            seq[(long)(fr0 + m) * 64 + n] = fmaxf((acc[nt][i] + bs) * sc + sh, 0.f);
        }
    }
}

// ---------------------------------------------------------------------------
// Stage 4a: LSTM input-gate GEMM: G[dir][512][256] = X @ Wih^T + bih + bhh
// grid = 64 (dir*32 + row-tile), block = 32
// ---------------------------------------------------------------------------
__global__ void __launch_bounds__(32)
lstm_gates_kernel(const float* __restrict__ X, int K,
                  const float* __restrict__ Wih0, const float* __restrict__ Wih1,
                  const float* __restrict__ bih0, const float* __restrict__ bhh0,
                  const float* __restrict__ bih1, const float* __restrict__ bhh1,
                  float* __restrict__ G)
{
    __shared__ float sB[32 * 256];   // pair layout
    __shared__ float sA[16 * 128];

    const int lane = threadIdx.x;
    const int mrow = lane & 15;
    const int kb = (lane >> 4) << 1;
    const int dir = blockIdx.x >> 5;
    const int r0 = (blockIdx.x & 31) * 16;
    const float* Wih = dir ? Wih1 : Wih0;
    const float* bi  = dir ? bih1 : bih0;
    const float* bh  = dir ? bhh1 : bhh0;

    v8f acc[16];
    for (int t = 0; t < 16; ++t) acc[t] = zero8();

    for (int i = lane; i < 16 * K; i += 32)
        sA[i] = X[(long)(r0 + i / K) * K + (i % K)];

    const int nkc = K / 32;
    for (int kc = 0; kc < nkc; ++kc) {
        for (int i = lane; i < 32 * 256; i += 32) {
            int k = i >> 8, n = i & 255;
            sB[BP(k, n, 256)] = Wih[(long)n * K + kc * 32 + k];
        }
        __syncthreads();
        for (int kk = 0; kk < 8; ++kk) {
            int k0b = kk * 4 + kb;
            int k0a = kc * 32 + k0b;
            v2f a = ld2(&sA[mrow * K + k0a]);
            for (int nt = 0; nt < 16; ++nt) {
                v2f b = ld2(&sB[((k0b >> 1) * 256 + nt * 16 + mrow) * 2]);
                acc[nt] = wmma4(a, b, acc[nt]);
            }
        }
        __syncthreads();
    }
    for (int nt = 0; nt < 16; ++nt) {
        int n = nt * 16 + mrow;
        float bias = bi[n] + bh[n];
#pragma unroll
        for (int i = 0; i < 8; ++i) {
            int m = i + ((lane >> 4) << 3);
            G[((long)dir * 512 + r0 + m) * 256 + n] = acc[nt][i] + bias;
        }
    }
}

// ---------------------------------------------------------------------------
// Stage 4b: LSTM recurrence. grid = 2 (dir), block = 256 (8 waves x 2 ntiles)
// Whh held in VGPRs (pre-swizzled WMMA B operands) across all 64 timesteps.
// ---------------------------------------------------------------------------
__global__ void __launch_bounds__(256)
lstm_recur_kernel(const float* __restrict__ G,
                  const float* __restrict__ Whh0, const float* __restrict__ Whh1,
                  const float* __restrict__ h0, const float* __restrict__ c0, int sIdx0,
                  float* __restrict__ out, int outStride,
                  float* __restrict__ hn, float* __restrict__ cn)
{
    __shared__ float sH[16][64];   // rows 8..15 stay zero (M padding)
    __shared__ float sC[8][64];
    __shared__ float sG[8][256];

    const int tid = threadIdx.x, lane = tid & 31, wave = tid >> 5;
    const int mrow = lane & 15;
    const int kb = (lane >> 4) << 1;
    const int dir = blockIdx.x;
    const float* Whh = dir ? Whh1 : Whh0;

    // persistent B operand registers: 2 N-tiles per wave, 16 K-steps
    v2f breg[2][16];
#pragma unroll
    for (int nt = 0; nt < 2; ++nt) {
        int n = (wave * 2 + nt) * 16 + mrow;
#pragma unroll
        for (int kk = 0; kk < 16; ++kk) {
            int k0 = kk * 4 + kb;
            breg[nt][kk] = ld2(&Whh[(long)n * 64 + k0]);
        }
    }

    for (int i = tid; i < 16 * 64; i += 256) (&sH[0][0])[i] = 0.f;
    __syncthreads();
    for (int i = tid; i < 8 * 64; i += 256) {
        int b = i >> 6, hh = i & 63;
        sH[b][hh] = h0[((long)(sIdx0 + dir) * 8 + b) * 64 + hh];
        sC[b][hh] = c0[((long)(sIdx0 + dir) * 8 + b) * 64 + hh];
    }
    __syncthreads();

    for (int step = 0; step < 64; ++step) {
        const int t = dir ? (63 - step) : step;
        for (int nt = 0; nt < 2; ++nt) {
            int ng = (wave * 2 + nt) * 16 + mrow;
            v8f c;
#pragma unroll
            for (int i = 0; i < 8; ++i) {
                int m = i + ((lane >> 4) << 3);
                c[i] = (m < 8) ? G[((long)dir * 512 + m * 64 + t) * 256 + ng] : 0.f;
            }
#pragma unroll
            for (int kk = 0; kk < 16; ++kk) {
                int k0 = kk * 4 + kb;
                v2f a = ld2(&sH[mrow][k0]);
                c = wmma4(a, breg[nt][kk], c);
            }
#pragma unroll
            for (int i = 0; i < 8; ++i) {
                int m = i + ((lane >> 4) << 3);
                if (m < 8) sG[m][ng] = c[i];
            }
        }
        __syncthreads();
        for (int i = tid; i < 8 * 64; i += 256) {
            int b = i >> 6, hh = i & 63;
            float gi = sG[b][hh], gf = sG[b][64 + hh], gg = sG[b][128 + hh], go = sG[b][192 + hh];
            float cc = sigf(gf) * sC[b][hh] + sigf(gi) * tanhf(gg);
            float hv = sigf(go) * tanhf(cc);
            sC[b][hh] = cc;
            sH[b][hh] = hv;
            out[((long)b * 64 + t) * outStride + dir * 64 + hh] = hv;
        }
        __syncthreads();
    }

    for (int i = tid; i < 8 * 64; i += 256) {
        int b = i >> 6, hh = i & 63;
        hn[(long)(sIdx0 + dir) * 512 + b * 64 + hh] = sH[b][hh];
        cn[(long)(sIdx0 + dir) * 512 + b * 64 + hh] = sC[b][hh];
    }
}

// ---------------------------------------------------------------------------
extern "C" void kernel_launch(void* const* d_in, const int* in_sizes, int n_in,
                              void* d_out, int out_size, void* d_ws, size_t ws_size,
                              hipStream_t stream) {
    const float* x  = (const float*)d_in[0];
    const float* h0 = (const float*)d_in[1];
    const float* c0 = (const float*)d_in[2];
    // point params: 3..16
    const float* W1  = (const float*)d_in[3];  const float* b1  = (const float*)d_in[4];
    const float* g1  = (const float*)d_in[5];  const float* be1 = (const float*)d_in[6];
    const float* W2  = (const float*)d_in[7];  const float* b2  = (const float*)d_in[8];
    const float* g2  = (const float*)d_in[9];  const float* be2 = (const float*)d_in[10];
    const float* W3  = (const float*)d_in[11]; const float* b3  = (const float*)d_in[12];
    const float* g3  = (const float*)d_in[13]; const float* be3 = (const float*)d_in[14];
    const float* Wa  = (const float*)d_in[15]; const float* ba  = (const float*)d_in[16];
    // voxel params: 17..28
    const float* K1  = (const float*)d_in[17];
    const float* vb1 = (const float*)d_in[18]; const float* vg1 = (const float*)d_in[19];
    const float* vbe1= (const float*)d_in[20];
    const float* VW2 = (const float*)d_in[21]; const float* vb2 = (const float*)d_in[22];
    const float* vg2 = (const float*)d_in[23]; const float* vbe2= (const float*)d_in[24];
    const float* VW3 = (const float*)d_in[25]; const float* vb3 = (const float*)d_in[26];
    const float* vg3 = (const float*)d_in[27]; const float* vbe3= (const float*)d_in[28];
    // lstm params: 29..52
    const float* Wih[6]; const float* Whh[6]; const float* bih[6]; const float* bhh[6];
    for (int i = 0; i < 6; ++i) {
        Wih[i] = (const float*)d_in[29 + 4 * i];
        Whh[i] = (const float*)d_in[30 + 4 * i];
        bih[i] = (const float*)d_in[31 + 4 * i];
        bhh[i] = (const float*)d_in[32 + 4 * i];
    }

    float* out  = (float*)d_out;                 // inp (8,64,128) = 65536
    float* attn = out + 65536;                   // (13824,8,1)   = 110592
    float* hn   = out + 65536 + 110592;          // (6,8,64)      = 3072
    float* cn   = hn + 3072;                     // (6,8,64)      = 3072

    float* ws  = (float*)d_ws;
    float* gp  = ws;                 // 13824*8*31 = 3,428,352
    float* vec = gp + 3428352;       // 13824*64   =   884,736
    float* seq = vec + 884736;       // 512*64     =    32,768
    float* l0o = seq + 32768;        // 8*64*128   =    65,536
    float* l1o = l0o + 65536;        // 8*64*128   =    65,536
    float* G   = l1o + 65536;        // 2*512*256  =   262,144

    group_kernel<<<512, 128, 0, stream>>>(x, gp);
    pointnet_kernel<<<1728, 128, 0, stream>>>(gp, W1, b1, g1, be1, W2, b2, g2, be2,
                                              W3, b3, g3, be3, Wa, ba, attn, vec);
    voxel_kernel<<<32, 32, 0, stream>>>(vec, K1, vb1, vg1, vbe1,
                                        VW2, vb2, vg2, vbe2, VW3, vb3, vg3, vbe3, seq);
    // layer 0
    lstm_gates_kernel<<<64, 32, 0, stream>>>(seq, 64, Wih[0], Wih[1],
                                             bih[0], bhh[0], bih[1], bhh[1], G);
    lstm_recur_kernel<<<2, 256, 0, stream>>>(G, Whh[0], Whh[1], h0, c0, 0, l0o, 128, hn, cn);
    // layer 1
    lstm_gates_kernel<<<64, 32, 0, stream>>>(l0o, 128, Wih[2], Wih[3],
                                             bih[2], bhh[2], bih[3], bhh[3], G);
    lstm_recur_kernel<<<2, 256, 0, stream>>>(G, Whh[2], Whh[3], h0, c0, 2, l1o, 128, hn, cn);
    // layer 2 -> final output
    lstm_gates_kernel<<<64, 32, 0, stream>>>(l1o, 128, Wih[4], Wih[5],
                                             bih[4], bhh[4], bih[5], bhh[5], G);
    lstm_recur_kernel<<<2, 256, 0, stream>>>(G, Whh[4], Whh[5], h0, c0, 4, out, 128, hn, cn);
}